// MainViewEncoder_32693291057234
// MI455X (gfx1250) — compile-verified
//
#include <hip/hip_runtime.h>

#define N_NODES 100000
#define N_EDGES 1600000
#define F 128

typedef __attribute__((ext_vector_type(2))) float v2f;
typedef __attribute__((ext_vector_type(8))) float v8f;

// ---------------- degree / normalization ----------------

__global__ void k_init_deg(unsigned* __restrict__ cnt) {
    int i = blockIdx.x * 256 + threadIdx.x;
    if (i < N_NODES) cnt[i] = 1u;  // self loop contributes 1 to deg
}

__global__ void k_deg(const int* __restrict__ dst, unsigned* __restrict__ cnt) {
    int e = blockIdx.x * 256 + threadIdx.x;
    if (e < N_EDGES) atomicAdd(&cnt[dst[e]], 1u);
}

__global__ void k_dinv(float* __restrict__ dinv) {
    int i = blockIdx.x * 256 + threadIdx.x;
    if (i < N_NODES) {
        unsigned c = ((const unsigned*)dinv)[i];
        dinv[i] = rsqrtf((float)c);   // deg >= 1 always (self loops)
    }
}

// ---------------- GEMM: h = x @ W  (fp32 WMMA 16x16x4) ----------------
// Block = 256 threads = 8 waves. Each wave computes a 16-row x 128-col strip.
// W (64KB) staged in LDS with a bank skew: row r stored with column rotate
// ((r&2)<<4) so lanes 0-15 (row kk) and lanes 16-31 (row kk+2) hit disjoint
// bank halves.
__global__ __launch_bounds__(256) void k_gemm(const float* __restrict__ x,
                                              const float* __restrict__ W,
                                              const float* __restrict__ dinv,
                                              float* __restrict__ h,
                                              float* __restrict__ out) {
    __shared__ float sW[F * F];  // 64 KB
    int tid = threadIdx.x;
    for (int t = tid; t < F * F; t += 256) {
        int r = t >> 7, c = t & 127;
        sW[(r << 7) + ((c + ((r & 2) << 4)) & 127)] = W[t];
    }
    __syncthreads();

    int lane = tid & 31;
    int wave = tid >> 5;
    int half = lane >> 4;     // 0: K=0,1  1: K=2,3 (A/B frag split)
    int lm   = lane & 15;

    int rowbase = blockIdx.x * 128 + wave * 16;
    if (rowbase >= N_NODES) return;   // uniform per wave: EXEC stays all-ones

    const float* __restrict__ xr = x + (size_t)(rowbase + lm) * F;

    v8f acc[8] = {};

    for (int k = 0; k < F; k += 4) {
        int kk = k + half * 2;
        v2f a = *(const v2f*)(xr + kk);               // A: row=rowbase+lm, K=kk,kk+1
        int sh0 = ((kk & 2) << 4);
        int sh1 = (((kk + 1) & 2) << 4);
#pragma unroll
        for (int j = 0; j < 8; ++j) {
            v2f b;
            int col = j * 16 + lm;
            b.x = sW[(kk << 7)       + ((col + sh0) & 127)];   // B: K=kk,   N=col
            b.y = sW[((kk + 1) << 7) + ((col + sh1) & 127)];   // B: K=kk+1, N=col
            acc[j] = __builtin_amdgcn_wmma_f32_16x16x4_f32(
                false, a, false, b, (short)0, acc[j], false, false);
        }
    }

    // Store h, and fused self-loop init: out = h * dinv^2
#pragma unroll
    for (int r = 0; r < 8; ++r) {
        int row = rowbase + half * 8 + r;   // C/D layout: VGPR r -> M=r (lanes<16) / M=8+r
        float dv = dinv[row];
        float d2 = dv * dv;
        size_t base = (size_t)row * F;
#pragma unroll
        for (int j = 0; j < 8; ++j) {
            float v = acc[j][r];
            int col = j * 16 + lm;
            h[base + col]   = v;
            out[base + col] = v * d2;
        }
    }
}

// ---------------- edge scatter: out[dst] += h[src] * dinv[src]*dinv[dst] ----
// One wave per edge; each lane handles 4 features (float4 gather, 4 fadd atomics).
__global__ __launch_bounds__(256) void k_edges(const int* __restrict__ src,
                                               const int* __restrict__ dst,
                                               const float* __restrict__ h,
                                               const float* __restrict__ dinv,
                                               float* __restrict__ out) {
    int gid  = blockIdx.x * 256 + threadIdx.x;
    int e    = gid >> 5;
    int lane = gid & 31;
    if (e >= N_EDGES) return;
    int s = src[e];
    int d = dst[e];
    float w = dinv[s] * dinv[d];
    float4 hv = *(const float4*)(h + (size_t)s * F + lane * 4);
    float* op = out + (size_t)d * F + lane * 4;
    unsafeAtomicAdd(op + 0, hv.x * w);
    unsafeAtomicAdd(op + 1, hv.y * w);
    unsafeAtomicAdd(op + 2, hv.z * w);
    unsafeAtomicAdd(op + 3, hv.w * w);
}

// ---------------- bias + relu ----------------
__global__ void k_bias_relu(float* __restrict__ out, const float* __restrict__ b) {
    int idx = blockIdx.x * 256 + threadIdx.x;   // over N*F/4 float4s
    if (idx >= (N_NODES * F) / 4) return;
    int f4 = idx & 31;
    float4 bb = *(const float4*)(b + f4 * 4);
    float4 v  = *(float4*)(out + (size_t)idx * 4);
    v.x = fmaxf(v.x + bb.x, 0.0f);
    v.y = fmaxf(v.y + bb.y, 0.0f);
    v.z = fmaxf(v.z + bb.z, 0.0f);
    v.w = fmaxf(v.w + bb.w, 0.0f);
    *(float4*)(out + (size_t)idx * 4) = v;
}

// ---------------- launcher ----------------
extern "C" void kernel_launch(void* const* d_in, const int* in_sizes, int n_in,
                              void* d_out, int out_size, void* d_ws, size_t ws_size,
                              hipStream_t stream) {
    const float* x  = (const float*)d_in[0];
    const int*   ei = (const int*)d_in[1];     // [2, E] flat
    const float* W  = (const float*)d_in[2];
    const float* b  = (const float*)d_in[3];
    const int* src = ei;
    const int* dst = ei + N_EDGES;

    float* out = (float*)d_out;
    float* h    = (float*)d_ws;                                       // 51.2 MB
    float* dinv = (float*)((char*)d_ws + (size_t)N_NODES * F * 4);    // 400 KB
    unsigned* cnt = (unsigned*)dinv;

    k_init_deg<<<(N_NODES + 255) / 256, 256, 0, stream>>>(cnt);
    k_deg<<<(N_EDGES + 255) / 256, 256, 0, stream>>>(dst, cnt);
    k_dinv<<<(N_NODES + 255) / 256, 256, 0, stream>>>(dinv);
    k_gemm<<<(N_NODES + 127) / 128, 256, 0, stream>>>(x, W, dinv, h, out);
    k_edges<<<(N_EDGES * 32 + 255) / 256, 256, 0, stream>>>(src, dst, h, dinv, out);
    k_bias_relu<<<((N_NODES * F) / 4 + 255) / 256, 256, 0, stream>>>(out, b);
}